// DMANet_Detector_88021059764755
// MI455X (gfx1250) — compile-verified
//
#include <hip/hip_runtime.h>
#include <stdint.h>

// ---------------- problem constants (match reference) ----------------
static constexpr int kA = 131072;   // anchors
static constexpr int kC = 8;        // classes
static constexpr int kK = 2048;     // top-k per class
static constexpr int kH = 256;
static constexpr int kW = 320;
static constexpr float kConf = 0.99f;
// IoU threshold 0.5 folded into "3*inter - ai - aj > 0"

static constexpr int kWords = kK / 32;          // 64 mask words per row
static constexpr int kTilesI = kK / 16;         // 128 row tiles
static constexpr int kWavesPerClass = kTilesI * kWords; // 8192 = 2^13

// ---------------- workspace layout (bytes) ----------------
static constexpr size_t OFS_BOXES  = 0;                         // A * float4      = 2 MB
static constexpr size_t OFS_CNT    = OFS_BOXES  + (size_t)kA * 16;          // C u32
static constexpr size_t OFS_LIST   = OFS_CNT    + 256;          // C*K u64        = 128 KB
static constexpr size_t OFS_SBOX   = OFS_LIST   + (size_t)kC * kK * 8;      // C*K float4 = 256 KB
static constexpr size_t OFS_SSCORE = OFS_SBOX   + (size_t)kC * kK * 16;     // C*K f32
static constexpr size_t OFS_AREA   = OFS_SSCORE + (size_t)kC * kK * 4;      // C*K f32
static constexpr size_t OFS_MASK   = OFS_AREA   + (size_t)kC * kK * 4;      // C*K*64 u32 = 4 MB
static constexpr size_t OFS_KEEP   = OFS_MASK   + (size_t)kC * kK * kWords * 4; // C*64 u32
// total ~6.9 MB

typedef __attribute__((ext_vector_type(2))) float v2f;
typedef __attribute__((ext_vector_type(8))) float v8f;

__device__ __forceinline__ uint32_t ballot32(bool p) {
  return (uint32_t)__ballot(p);   // wave32: low 32 bits
}

// ---------------- 0) re-init mutable scratch (graph-replay safe) ----------------
__global__ void init_kernel(uint32_t* cnt, uint64_t* lists) {
  int t = blockIdx.x * blockDim.x + threadIdx.x;
  if (t < kC) cnt[t] = 0u;
  int total = kC * kK;
  int stride = gridDim.x * blockDim.x;
  for (int i = t; i < total; i += stride) lists[i] = 0ull;   // pad key = 0 (score 0)
}

// ---------------- 1) decode + clip + threshold-append ----------------
__global__ void decode_kernel(const float* __restrict__ cls,
                              const float* __restrict__ reg,
                              const float* __restrict__ anc,
                              float4* __restrict__ boxes,
                              uint32_t* __restrict__ cnt,
                              uint64_t* __restrict__ lists) {
  int a = blockIdx.x * blockDim.x + threadIdx.x;
  if (a >= kA) return;
  float x1 = anc[a * 4 + 0], y1 = anc[a * 4 + 1];
  float x2 = anc[a * 4 + 2], y2 = anc[a * 4 + 3];
  float wa = x2 - x1, ha = y2 - y1;
  float cxa = x1 + 0.5f * wa, cya = y1 + 0.5f * ha;
  float dx = reg[a * 4 + 0] * 0.1f, dy = reg[a * 4 + 1] * 0.1f;
  float dw = reg[a * 4 + 2] * 0.2f, dh = reg[a * 4 + 3] * 0.2f;
  float pcx = cxa + dx * wa, pcy = cya + dy * ha;
  float pw = expf(dw) * wa,  ph = expf(dh) * ha;
  float bx1 = fmaxf(pcx - 0.5f * pw, 0.0f);
  float by1 = fmaxf(pcy - 0.5f * ph, 0.0f);
  float bx2 = fminf(pcx + 0.5f * pw, (float)kW);
  float by2 = fminf(pcy + 0.5f * ph, (float)kH);
  boxes[a] = make_float4(bx1, by1, bx2, by2);

#pragma unroll
  for (int c = 0; c < kC; ++c) {
    float s = cls[a * kC + c];
    if (s > kConf) {
      uint32_t pos = atomicAdd(&cnt[c], 1u);
      if (pos < (uint32_t)kK) {
        // key: score bits (scores >= 0 so bits are order-preserving) ; ties -> smaller idx first
        uint64_t key = ((uint64_t)__float_as_uint(s) << 32) | (uint32_t)(~(uint32_t)a);
        lists[(size_t)c * kK + pos] = key;
      }
    }
  }
}

// ---------------- 2) per-class bitonic sort (desc) + gather boxes/areas ----------------
__global__ __launch_bounds__(1024) void sort_gather_kernel(
    const uint64_t* __restrict__ lists, const float4* __restrict__ boxes,
    float4* __restrict__ sbox, float* __restrict__ sscore, float* __restrict__ sarea) {
  __shared__ __align__(16) uint64_t keys[kK];
  int c = blockIdx.x;
  int tid = threadIdx.x;
  const uint64_t* src = lists + (size_t)c * kK;

  // Stage 16 KB of keys into LDS with the gfx1250 async LDS-DMA path (ASYNCcnt).
  {
#if defined(__gfx1250__)
    // generic "shared" pointers carry the LDS byte offset in their low 32 bits
    uint32_t ldsoff = (uint32_t)(uintptr_t)(&keys[tid * 2]);
    unsigned long long gaddr = (unsigned long long)(uintptr_t)(src + tid * 2);
    asm volatile("global_load_async_to_lds_b128 %0, %1, off"
                 :: "v"(ldsoff), "v"(gaddr) : "memory");
    asm volatile("s_wait_asynccnt 0" ::: "memory");
#else
    keys[tid * 2 + 0] = src[tid * 2 + 0];
    keys[tid * 2 + 1] = src[tid * 2 + 1];
#endif
  }
  __syncthreads();

  // bitonic sort, descending (pad keys are 0 and sink to the bottom)
  for (unsigned k = 2; k <= (unsigned)kK; k <<= 1) {
    for (unsigned j = k >> 1; j > 0; j >>= 1) {
      for (unsigned i = tid; i < (unsigned)kK; i += 1024u) {
        unsigned ixj = i ^ j;
        if (ixj > i) {
          uint64_t va = keys[i], vb = keys[ixj];
          bool descSeg = ((i & k) == 0);
          if (descSeg ? (va < vb) : (va > vb)) { keys[i] = vb; keys[ixj] = va; }
        }
      }
      __syncthreads();
    }
  }

  for (unsigned i = tid; i < (unsigned)kK; i += 1024u) {
    uint64_t key = keys[i];
    float s = __uint_as_float((uint32_t)(key >> 32));
    bool valid = s > kConf;
    uint32_t aidx = valid ? ~(uint32_t)key : 0u;
    float4 b = valid ? boxes[aidx] : make_float4(0.f, 0.f, 0.f, 0.f);
    size_t o = (size_t)c * kK + i;
    sbox[o] = b;
    sscore[o] = valid ? s : 0.0f;
    sarea[o] = fmaxf(b.z - b.x, 0.0f) * fmaxf(b.w - b.y, 0.0f);
  }
}

// ---------------- 3) suppression bit-matrix, WMMA-assisted ----------------
// One wave computes a 16-row x 32-col strip of mask bits for one class.
// iou > 0.5  <=>  3*inter - area_i - area_j > 0.
// The rank-2 bias (-area_i - area_j) over a 16x16 tile is D = A x B + C with
//   A(16x4) = [-area_i | -1 | 0 | 0],  B(4x16) = [1 ; area_j ; 0 ; 0],  C = 3*inter.
__global__ __launch_bounds__(256) void mask_kernel(const float4* __restrict__ sbox,
                                                   const float* __restrict__ sarea,
                                                   uint32_t* __restrict__ mask) {
  int wave = blockIdx.x * (256 / 32) + ((int)threadIdx.x >> 5);
  int lane = (int)threadIdx.x & 31;
  int wordJ = wave & (kWords - 1);
  int tileI = (wave >> 6) & (kTilesI - 1);
  int c = wave >> 13;                       // kWavesPerClass = 2^13

  const float4* B = sbox + (size_t)c * kK;
  const float* AR = sarea + (size_t)c * kK;

  int n = lane & 15;
  bool hi = lane >= 16;
  int j0 = wordJ * 32 + n, j1 = j0 + 16;
  float4 bj0 = B[j0], bj1 = B[j1];
  float aj0 = AR[j0], aj1 = AR[j1];
  int i0 = tileI * 16 + (hi ? 8 : 0);       // per-lane row base (C/D layout: M = v (+8 for hi half))

  v8f c0, c1;
#pragma unroll
  for (int v = 0; v < 8; ++v) {
    float4 bi = B[i0 + v];
    float ix1 = fmaxf(bi.x, bj0.x), iy1 = fmaxf(bi.y, bj0.y);
    float ix2 = fminf(bi.z, bj0.z), iy2 = fminf(bi.w, bj0.w);
    c0[v] = 3.0f * (fmaxf(ix2 - ix1, 0.0f) * fmaxf(iy2 - iy1, 0.0f));
    ix1 = fmaxf(bi.x, bj1.x); iy1 = fmaxf(bi.y, bj1.y);
    ix2 = fminf(bi.z, bj1.z); iy2 = fminf(bi.w, bj1.w);
    c1[v] = 3.0f * (fmaxf(ix2 - ix1, 0.0f) * fmaxf(iy2 - iy1, 0.0f));
  }

  // A layout (16x4 f32): lanes 0-15 hold M=lane, vgpr0=K0, vgpr1=K1; lanes 16-31 hold K2,K3 (zero here).
  // B layout (4x16 f32): lanes 0-15 vgpr0=row K0, vgpr1=row K1; lanes 16-31 rows K2,K3 (zero here).
  float aIl = AR[tileI * 16 + (lane & 15)];
  v2f av, b0, b1;
  av.x = hi ? 0.0f : -aIl;  av.y = hi ? 0.0f : -1.0f;
  b0.x = hi ? 0.0f : 1.0f;  b0.y = hi ? 0.0f : aj0;
  b1.x = hi ? 0.0f : 1.0f;  b1.y = hi ? 0.0f : aj1;

#if __has_builtin(__builtin_amdgcn_wmma_f32_16x16x4_f32)
  v8f d0 = __builtin_amdgcn_wmma_f32_16x16x4_f32(false, av, false, b0, (short)0, c0, false, false);
  v8f d1 = __builtin_amdgcn_wmma_f32_16x16x4_f32(false, av, false, b1, (short)0, c1, false, false);
#else
  v8f d0, d1;
#pragma unroll
  for (int v = 0; v < 8; ++v) {
    d0[v] = c0[v] - AR[i0 + v] - aj0;
    d1[v] = c1[v] - AR[i0 + v] - aj1;
  }
#endif

  uint32_t myword = 0u;
#pragma unroll
  for (int v = 0; v < 8; ++v) {
    int iv = i0 + v;  // this lane's global row for accumulator register v
    uint32_t m0 = ballot32(d0[v] > 0.0f && j0 > iv);
    uint32_t m1 = ballot32(d1[v] > 0.0f && j1 > iv);
    uint32_t wlow  = (m0 & 0xFFFFu) | ((m1 & 0xFFFFu) << 16);  // row tileI*16+v
    uint32_t whigh = (m0 >> 16) | (m1 & 0xFFFF0000u);          // row tileI*16+v+8
    if (lane == v)     myword = wlow;
    if (lane == v + 8) myword = whigh;
  }
  if (lane < 16)
    mask[((size_t)c * kK + tileI * 16 + lane) * kWords + wordJ] = myword;
}

// ---------------- 4) serial greedy scan, one wave per class ----------------
__global__ __launch_bounds__(32) void scan_kernel(const float* __restrict__ sscore,
                                                  const uint32_t* __restrict__ mask,
                                                  uint32_t* __restrict__ keep) {
  int c = blockIdx.x;
  int lane = (int)threadIdx.x;   // wave32: lane owns removed words lane and lane+32
  const float* S = sscore + (size_t)c * kK;
  const uint32_t* M = mask + (size_t)c * kK * kWords;
  uint32_t rem_lo = 0u, rem_hi = 0u, keep_lo = 0u, keep_hi = 0u;
  for (int i = 0; i < kK; ++i) {
    int widx = i >> 5, bit = i & 31;
    uint32_t wlo = __shfl(rem_lo, widx & 31, 32);
    uint32_t whi = __shfl(rem_hi, widx & 31, 32);
    uint32_t wv = (widx < 32) ? wlo : whi;
    bool removed = (wv >> bit) & 1u;
    bool valid = S[i] > kConf;           // uniform across the wave
    if (valid && !removed) {
      if ((widx & 31) == lane) {
        if (widx < 32) keep_lo |= (1u << bit); else keep_hi |= (1u << bit);
      }
      rem_lo |= M[(size_t)i * kWords + lane];
      rem_hi |= M[(size_t)i * kWords + 32 + lane];
    }
  }
  keep[c * kWords + lane] = keep_lo;
  keep[c * kWords + 32 + lane] = keep_hi;
}

// ---------------- 5) emit [C*K, 6] rows ----------------
__global__ void output_kernel(const float4* __restrict__ sbox,
                              const float* __restrict__ sscore,
                              const uint32_t* __restrict__ keep,
                              float* __restrict__ out) {
  int t = blockIdx.x * blockDim.x + threadIdx.x;
  if (t >= kC * kK) return;
  int c = t / kK, k = t - c * kK;
  bool kept = (keep[c * kWords + (k >> 5)] >> (k & 31)) & 1u;
  float4 b = sbox[t];
  float s = sscore[t];
  float* o = out + (size_t)t * 6;
  o[0] = kept ? b.x : 0.0f;
  o[1] = kept ? b.y : 0.0f;
  o[2] = kept ? b.z : 0.0f;
  o[3] = kept ? b.w : 0.0f;
  o[4] = kept ? s : 0.0f;
  o[5] = kept ? (float)c : 0.0f;
}

extern "C" void kernel_launch(void* const* d_in, const int* in_sizes, int n_in,
                              void* d_out, int out_size, void* d_ws, size_t ws_size,
                              hipStream_t stream) {
  const float* cls = (const float*)d_in[0];   // (1, A, C)
  const float* reg = (const float*)d_in[1];   // (1, A, 4)
  const float* anc = (const float*)d_in[2];   // (1, A, 4)
  // d_in[3] = img_batch, only its static shape (H=256, W=320) is used
  (void)in_sizes; (void)n_in; (void)out_size; (void)ws_size;

  char* ws = (char*)d_ws;
  float4*   boxes  = (float4*)  (ws + OFS_BOXES);
  uint32_t* cnt    = (uint32_t*)(ws + OFS_CNT);
  uint64_t* lists  = (uint64_t*)(ws + OFS_LIST);
  float4*   sbox   = (float4*)  (ws + OFS_SBOX);
  float*    sscore = (float*)   (ws + OFS_SSCORE);
  float*    sarea  = (float*)   (ws + OFS_AREA);
  uint32_t* mask   = (uint32_t*)(ws + OFS_MASK);
  uint32_t* keep   = (uint32_t*)(ws + OFS_KEEP);
  float*    out    = (float*)d_out;

  init_kernel<<<64, 256, 0, stream>>>(cnt, lists);
  decode_kernel<<<kA / 256, 256, 0, stream>>>(cls, reg, anc, boxes, cnt, lists);
  sort_gather_kernel<<<kC, 1024, 0, stream>>>(lists, boxes, sbox, sscore, sarea);
  mask_kernel<<<(kC * kWavesPerClass) / 8, 256, 0, stream>>>(sbox, sarea, mask);
  scan_kernel<<<kC, 32, 0, stream>>>(sscore, mask, keep);
  output_kernel<<<(kC * kK) / 256, 256, 0, stream>>>(sbox, sscore, keep, out);
}